// zAttention_34746285425377
// MI455X (gfx1250) — compile-verified
//
#include <hip/hip_runtime.h>
#include <math.h>

typedef __attribute__((ext_vector_type(2))) float v2f;
typedef __attribute__((ext_vector_type(8))) float v8f;

#define NSEQ  64
#define NHEAD 32
#define HDIM  128
#define NKVH  8
#define GQA   4
#define BS    16
#define MAXB  128
#define WAVES 4

static __device__ __forceinline__ v8f wmma4(v2f a, v2f b, v8f c) {
  // V_WMMA_F32_16X16X4_F32 : D(16x16) = A(16x4) * B(4x16) + C
  return __builtin_amdgcn_wmma_f32_16x16x4_f32(
      /*neg_a=*/false, a, /*neg_b=*/false, b,
      /*c_mod=*/(short)0, c, /*reuse_a=*/false, /*reuse_b=*/false);
}

// One async 512B-per-wave copy: 16B per lane, instruction offset advances BOTH
// the global source and the LDS destination (ISA 10.7/15.18.3 semantics).
#define ACP(ldsoff, g64, OFFLIT)                                          \
  asm volatile("global_load_async_to_lds_b128 %0, %1, off offset:" OFFLIT \
               ::"v"(ldsoff), "v"(g64) : "memory")

// Stage a 16x128 f32 K tile and V tile (8KB each) asynchronously into LDS.
static __device__ __forceinline__ void stage_kv_async(const float* gk, const float* gv,
                                                      const float* lk, const float* lv,
                                                      int lane) {
  const unsigned int lko = (unsigned int)(size_t)lk + lane * 16;
  const unsigned int lvo = (unsigned int)(size_t)lv + lane * 16;
  const unsigned long long gk64 = (unsigned long long)(size_t)gk + lane * 16;
  const unsigned long long gv64 = (unsigned long long)(size_t)gv + lane * 16;
  ACP(lko, gk64, "0");    ACP(lvo, gv64, "0");
  ACP(lko, gk64, "512");  ACP(lvo, gv64, "512");
  ACP(lko, gk64, "1024"); ACP(lvo, gv64, "1024");
  ACP(lko, gk64, "1536"); ACP(lvo, gv64, "1536");
  ACP(lko, gk64, "2048"); ACP(lvo, gv64, "2048");
  ACP(lko, gk64, "2560"); ACP(lvo, gv64, "2560");
  ACP(lko, gk64, "3072"); ACP(lvo, gv64, "3072");
  ACP(lko, gk64, "3584"); ACP(lvo, gv64, "3584");
  ACP(lko, gk64, "4096"); ACP(lvo, gv64, "4096");
  ACP(lko, gk64, "4608"); ACP(lvo, gv64, "4608");
  ACP(lko, gk64, "5120"); ACP(lvo, gv64, "5120");
  ACP(lko, gk64, "5632"); ACP(lvo, gv64, "5632");
  ACP(lko, gk64, "6144"); ACP(lvo, gv64, "6144");
  ACP(lko, gk64, "6656"); ACP(lvo, gv64, "6656");
  ACP(lko, gk64, "7168"); ACP(lvo, gv64, "7168");
  ACP(lko, gk64, "7680"); ACP(lvo, gv64, "7680");
}

extern "C" __global__ void __launch_bounds__(WAVES * 32)
pa_decode_kernel(const float* __restrict__ q, const float* __restrict__ kc,
                 const float* __restrict__ vc, const int* __restrict__ bt,
                 const int* __restrict__ cl, float* __restrict__ out) {
  // LDS: Q tile 16x128 | per wave: K0 K1 V0 V1 (16x128 each) + P 16x16
  __shared__ float lds[2048 + WAVES * (4 * 2048 + 256)];

  const int s    = blockIdx.x >> 3;
  const int h    = blockIdx.x & 7;
  const int tid  = threadIdx.x;
  const int w    = tid >> 5;
  const int lane = tid & 31;
  const int lr   = lane & 15;
  const int hf   = lane >> 4;

  const float scale = 0.08838834764831845f;  // 1/sqrt(128)

  float* lq    = lds;
  float* wbase = lds + 2048 + w * (4 * 2048 + 256);
  float* lp    = wbase + 8192;

  // ---- init shared Q tile (pre-scaled, rows 4..15 zero) + zero P tile ----
  for (int i = tid; i < 16 * HDIM; i += WAVES * 32) {
    const int row = i >> 7, d = i & 127;
    float val = 0.0f;
    if (row < GQA)
      val = q[((size_t)s * NHEAD + h * GQA + row) * HDIM + d] * scale;
    lq[i] = val;
  }
  for (int i = lane; i < 256; i += 32) lp[i] = 0.0f;
  __syncthreads();

  const int ctx = cl[s];
  const int nb  = (ctx + BS - 1) / BS;

  const v8f vzero = {0.f, 0.f, 0.f, 0.f, 0.f, 0.f, 0.f, 0.f};
  v8f acc[8];
#pragma unroll
  for (int t = 0; t < 8; ++t) acc[t] = vzero;
  float mrow[GQA], lrow[GQA];
#pragma unroll
  for (int g = 0; g < GQA; ++g) { mrow[g] = -__builtin_inff(); lrow[g] = 0.0f; }

  // ---- prologue: async-stage this wave's first block into buffer 0 ----
  if (w < nb) {
    const int bid0 = bt[(size_t)s * MAXB + w];
    stage_kv_async(kc + ((size_t)bid0 * NKVH + h) * (BS * HDIM),
                   vc + ((size_t)bid0 * NKVH + h) * (BS * HDIM),
                   wbase, wbase + 4096, lane);
  }

  int bufoff = 0;  // 0 or 2048 floats: double-buffer select (pure arithmetic,
                   // keeps LDS addrspace inference intact -> ds_load, not flat)
  for (int j = w; j < nb; j += WAVES) {
    float* lk = wbase + bufoff;
    float* lv = wbase + 4096 + bufoff;

    // issue next block's async copies into the other buffer, then wait for
    // the current block's 32 copies (async loads complete in order).
    if (j + WAVES < nb) {
      const int nbid = bt[(size_t)s * MAXB + j + WAVES];
      stage_kv_async(kc + ((size_t)nbid * NKVH + h) * (BS * HDIM),
                     vc + ((size_t)nbid * NKVH + h) * (BS * HDIM),
                     wbase + (bufoff ^ 2048), wbase + 4096 + (bufoff ^ 2048), lane);
      asm volatile("s_wait_asynccnt 32" ::: "memory");
    } else {
      asm volatile("s_wait_asynccnt 0" ::: "memory");
    }

    // ---- scores S(16x16) = Qtile * K^T via 32 chained f32 WMMAs ----
    v8f sc = vzero;
    {
      const v2f* qa = (const v2f*)(lq + lr * HDIM + 2 * hf);
      const v2f* kf = (const v2f*)(lk + lr * HDIM + 2 * hf);
#pragma unroll
      for (int c = 0; c < 32; ++c) sc = wmma4(qa[2 * c], kf[2 * c], sc);
    }

    // mask invalid tokens of the (possibly partial) final block
    if (j * BS + lr >= ctx) {
#pragma unroll
      for (int r = 0; r < 8; ++r) sc[r] = -__builtin_inff();
    }

    // ---- online softmax per q-head (row g = vgpr g, lanes 0..15) ----
#pragma unroll
    for (int g = 0; g < GQA; ++g) {
      float sg = sc[g];
      float bm = sg;
      bm = fmaxf(bm, __shfl_xor(bm, 1, 32));
      bm = fmaxf(bm, __shfl_xor(bm, 2, 32));
      bm = fmaxf(bm, __shfl_xor(bm, 4, 32));
      bm = fmaxf(bm, __shfl_xor(bm, 8, 32));
      const float mn = fmaxf(mrow[g], bm);
      const float al = __expf(mrow[g] - mn);
      const float p  = __expf(sg - mn);
      float bs = p;
      bs += __shfl_xor(bs, 1, 32);
      bs += __shfl_xor(bs, 2, 32);
      bs += __shfl_xor(bs, 4, 32);
      bs += __shfl_xor(bs, 8, 32);
      lrow[g] = lrow[g] * al + bs;
      mrow[g] = mn;
#pragma unroll
      for (int t = 0; t < 8; ++t) acc[t][g] *= al;
      if (lane < 16) lp[g * 16 + lane] = p;  // rows 4..15 stay zero
    }
    asm volatile("s_wait_dscnt 0" ::: "memory");  // cross-lane P visibility

    // ---- O(16x128) += P(16x16) * V(16x128): 4 K-chunks x 8 N-tiles ----
#pragma unroll
    for (int cc = 0; cc < 4; ++cc) {
      const v2f a = *(const v2f*)(lp + lr * 16 + 4 * cc + 2 * hf);
      const int t0 = 4 * cc + 2 * hf;
#pragma unroll
      for (int dt = 0; dt < 8; ++dt) {
        v2f b;
        b.x = lv[t0 * HDIM + dt * 16 + lr];
        b.y = lv[(t0 + 1) * HDIM + dt * 16 + lr];
        acc[dt] = wmma4(a, b, acc[dt]);
      }
    }
    bufoff ^= 2048;
  }

  // ---- cross-wave flash combine (overlay wave0's staging buffers) ----
  __syncthreads();
  float* opart = lds + 2048;         // [w][g][d] : 4*4*128 floats
  float* ml    = lds + 2048 + 2048;  // [w][g][{m,l}] : 32 floats

  if (lane < 16) {
#pragma unroll
    for (int g = 0; g < GQA; ++g) {
#pragma unroll
      for (int dt = 0; dt < 8; ++dt)
        opart[((w * GQA + g) << 7) + dt * 16 + lr] = acc[dt][g];
      if (lr == 0) {
        ml[(w * GQA + g) * 2 + 0] = mrow[g];
        ml[(w * GQA + g) * 2 + 1] = lrow[g];
      }
    }
  }
  __syncthreads();

  {
    const int g = tid >> 5;   // 0..3
    const int i = tid & 31;   // d = 4*i
    const float m0 = ml[(0 * GQA + g) * 2], m1 = ml[(1 * GQA + g) * 2];
    const float m2 = ml[(2 * GQA + g) * 2], m3 = ml[(3 * GQA + g) * 2];
    const float ms = fmaxf(fmaxf(m0, m1), fmaxf(m2, m3));
    const float c0 = __expf(m0 - ms), c1 = __expf(m1 - ms);
    const float c2 = __expf(m2 - ms), c3 = __expf(m3 - ms);
    const float ls = c0 * ml[(0 * GQA + g) * 2 + 1] + c1 * ml[(1 * GQA + g) * 2 + 1] +
                     c2 * ml[(2 * GQA + g) * 2 + 1] + c3 * ml[(3 * GQA + g) * 2 + 1];
    const float4* o4 = (const float4*)opart;
    const float4 o0 = o4[((0 * GQA + g) << 5) + i];
    const float4 o1 = o4[((1 * GQA + g) << 5) + i];
    const float4 o2 = o4[((2 * GQA + g) << 5) + i];
    const float4 o3 = o4[((3 * GQA + g) << 5) + i];
    const float inv = 1.0f / ls;
    float4 r;
    r.x = (c0 * o0.x + c1 * o1.x + c2 * o2.x + c3 * o3.x) * inv;
    r.y = (c0 * o0.y + c1 * o1.y + c2 * o2.y + c3 * o3.y) * inv;
    r.z = (c0 * o0.z + c1 * o1.z + c2 * o2.z + c3 * o3.z) * inv;
    r.w = (c0 * o0.w + c1 * o1.w + c2 * o2.w + c3 * o3.w) * inv;
    float* op = out + ((size_t)s * NHEAD + h * GQA + g) * HDIM + 4 * i;
    *(float4*)op = r;
  }
}

extern "C" void kernel_launch(void* const* d_in, const int* in_sizes, int n_in,
                              void* d_out, int out_size, void* d_ws, size_t ws_size,
                              hipStream_t stream) {
  (void)in_sizes; (void)n_in; (void)d_ws; (void)ws_size; (void)out_size;
  const float* q  = (const float*)d_in[0];
  const float* kc = (const float*)d_in[1];
  const float* vc = (const float*)d_in[2];
  const int*   bt = (const int*)d_in[3];
  const int*   cl = (const int*)d_in[4];
  float* outp = (float*)d_out;
  pa_decode_kernel<<<dim3(NSEQ * NKVH), dim3(WAVES * 32), 0, stream>>>(
      q, kc, vc, bt, cl, outp);
}